// LatentDimLSTM_85581518340210
// MI455X (gfx1250) — compile-verified
//
#include <hip/hip_runtime.h>

typedef __attribute__((ext_vector_type(2))) float v2f;
typedef __attribute__((ext_vector_type(8))) float v8f;

namespace {
constexpr int kHid     = 50;            // LSTM hidden size
constexpr int kD       = 64;            // input / output feature dim
constexpr int kTin     = 512;           // warm-up steps
constexpr int kTar     = 256;           // autoregressive steps
constexpr int kTout    = kTin + kTar;   // 768
constexpr int kBS      = 1024;
constexpr int kMT      = 16;            // batch rows per workgroup (one WMMA M-tile)
constexpr int kThreads = 512;           // 16 wave32s
constexpr int kKP      = 128;           // padded K storage: [x(64) | h(50->64)]
constexpr int kNP      = 256;           // padded gate cols: 4 gates * 64
constexpr int kKS      = 29;            // gate k-steps: K=116 covers real K=114
constexpr int kWRows   = 2 * kKS;       // 58 packed k-pair rows in weight LDS
constexpr int kKSo     = 13;            // proj k-steps: K=52 covers real K=50
constexpr int kWoRows  = 2 * kKSo;      // 26
}

__device__ __forceinline__ float sigf(float v) { return 1.0f / (1.0f + __expf(-v)); }
__device__ __forceinline__ float tanh_fast(float v) {
  v = fminf(fmaxf(v, -20.0f), 20.0f);
  float e = __expf(2.0f * v);
  return (e - 1.0f) / (e + 1.0f);
}

__global__ __launch_bounds__(kThreads)
void lstm_persistent_kernel(const float* __restrict__ x,
                            const float* __restrict__ Wih0, const float* __restrict__ Whh0,
                            const float* __restrict__ bih0, const float* __restrict__ bhh0,
                            const float* __restrict__ Wih1, const float* __restrict__ Whh1,
                            const float* __restrict__ bih1, const float* __restrict__ bhh1,
                            const float* __restrict__ Wout, const float* __restrict__ bout,
                            float* __restrict__ out)
{
  // ---- LDS: weights (transposed, zero-padded, packed in k-pairs) + state ----
  __shared__ float2 sW0[kWRows][kNP];    // 116 KB  W0^T[k][n], n = gate*64+c
  __shared__ float2 sW1[kWRows][kNP];    // 116 KB  W1^T[k][n]
  __shared__ float2 sWo[kWoRows][kD];    //  13 KB  Wout^T[k][n]
  __shared__ float  sXH0[kMT][kKP];      //   8 KB  [x_t | h0] per batch row
  __shared__ float  sHH1[kMT][kKP];      //   8 KB  [h0  | h1]
  __shared__ float  sG[kMT][kNP];        //  16 KB  raw gates / proj scratch
  __shared__ float  sC0[kMT][kD];        //   4 KB  c-state layer 0
  __shared__ float  sC1[kMT][kD];        //   4 KB  c-state layer 1
  __shared__ float  sB0[kNP];            //   1 KB  bih0+bhh0 (padded gate layout)
  __shared__ float  sB1[kNP];            //   1 KB
  __shared__ float  sBo[kD];             // 256 B
  __shared__ float  sXst[2][kMT][kD];    //   8 KB  async x double-buffer

  const int tid  = threadIdx.x;
  const int lane = tid & 31;
  const int wv   = tid >> 5;          // wave id 0..15
  const int half = lane >> 4;         // 0 | 1
  const int rA   = lane & 15;         // A-matrix row within 16-row tile
  const int nB   = (wv << 4) + (lane & 15);  // B/C column owned by this lane
  const int bm   = blockIdx.x * kMT;  // first batch row of this workgroup

  // one b64 (2 floats) of the x tile per thread, for async staging
  const int xr = tid >> 5;            // 0..15
  const int xc = (tid & 31) * 2;      // 0..62

  // async prefetch of x[:, t, :] tile into sXst[buf] (CDNA5 ASYNCcnt path)
  auto stage_issue = [&](int t, int buf) {
    const float* gp = &x[((long)(bm + xr) * kTin + t) * kD + xc];
    unsigned ldsa = (unsigned)(unsigned long long)&sXst[buf][xr][xc];
    unsigned long long ga = (unsigned long long)gp;
    asm volatile("global_load_async_to_lds_b64 %0, %1, off"
                 :: "v"(ldsa), "v"(ga) : "memory");
  };

  // kick off prefetch of t=0 immediately; it overlaps the weight staging below
  stage_issue(0, 0);

  // ---------------- one-time weight / bias staging ----------------
  // Layer 0: K = [Wih0 k<64 | Whh0 k-64<50 | 0], N = gate*64 + c (c<50 real)
  for (int i = tid; i < kWRows * kNP; i += kThreads) {
    int kp = i / kNP, n = i - kp * kNP;
    int g = n >> 6, c = n & 63;
    int row = g * kHid + c;
    float2 v; v.x = 0.0f; v.y = 0.0f;
    if (c < kHid) {
      int k0 = kp * 2, k1 = k0 + 1;
      v.x = (k0 < kD) ? Wih0[row * kD + k0]
                      : (((k0 - kD) < kHid) ? Whh0[row * kHid + (k0 - kD)] : 0.0f);
      v.y = (k1 < kD) ? Wih0[row * kD + k1]
                      : (((k1 - kD) < kHid) ? Whh0[row * kHid + (k1 - kD)] : 0.0f);
    }
    sW0[kp][n] = v;
  }
  // Layer 1: K = [Wih1 k<50 | 0 | Whh1 k-64<50 | 0]
  for (int i = tid; i < kWRows * kNP; i += kThreads) {
    int kp = i / kNP, n = i - kp * kNP;
    int g = n >> 6, c = n & 63;
    int row = g * kHid + c;
    float2 v; v.x = 0.0f; v.y = 0.0f;
    if (c < kHid) {
      int k0 = kp * 2, k1 = k0 + 1;
      v.x = (k0 < kHid) ? Wih1[row * kHid + k0]
                        : ((k0 >= kD && (k0 - kD) < kHid) ? Whh1[row * kHid + (k0 - kD)] : 0.0f);
      v.y = (k1 < kHid) ? Wih1[row * kHid + k1]
                        : ((k1 >= kD && (k1 - kD) < kHid) ? Whh1[row * kHid + (k1 - kD)] : 0.0f);
    }
    sW1[kp][n] = v;
  }
  // Output projection: Wout is [64][50]; K = h1 (50->52), N = 64
  for (int i = tid; i < kWoRows * kD; i += kThreads) {
    int kp = i >> 6, n = i & 63;
    int k0 = kp * 2, k1 = k0 + 1;
    float2 v;
    v.x = (k0 < kHid) ? Wout[n * kHid + k0] : 0.0f;
    v.y = (k1 < kHid) ? Wout[n * kHid + k1] : 0.0f;
    sWo[kp][n] = v;
  }
  // Fused biases in padded gate layout
  for (int n = tid; n < kNP; n += kThreads) {
    int g = n >> 6, c = n & 63;
    sB0[n] = (c < kHid) ? (bih0[g * kHid + c] + bhh0[g * kHid + c]) : 0.0f;
    sB1[n] = (c < kHid) ? (bih1[g * kHid + c] + bhh1[g * kHid + c]) : 0.0f;
  }
  for (int n = tid; n < kD; n += kThreads) sBo[n] = bout[n];
  // Zero-init states (padding columns stay zero forever)
  for (int i = tid; i < kMT * kKP; i += kThreads) {
    sXH0[i / kKP][i % kKP] = 0.0f;
    sHH1[i / kKP][i % kKP] = 0.0f;
  }
  for (int i = tid; i < kMT * kD; i += kThreads) {
    sC0[i >> 6][i & 63] = 0.0f;
    sC1[i >> 6][i & 63] = 0.0f;
  }
  __syncthreads();

  // ---------------- per-step building blocks ----------------
  // gates[16 x 256] = in[16 x ~116] @ W^T ; one N-tile (16 cols) per wave.
  // Two independent accumulator chains halve the serial WMMA dependency depth.
  auto gates_wmma = [&](const float (*in)[kKP], const float2 (*W)[kNP]) {
    v8f acc0 = {0.f, 0.f, 0.f, 0.f, 0.f, 0.f, 0.f, 0.f};
    v8f acc1 = {0.f, 0.f, 0.f, 0.f, 0.f, 0.f, 0.f, 0.f};
#pragma unroll
    for (int ks = 0; ks < kKS; ++ks) {
      // A 16x4 f32: lanes 0-15 hold K=k0,k0+1; lanes 16-31 hold K=k0+2,k0+3
      const float* ap = &in[rA][4 * ks + 2 * half];
      v2f a; a.x = ap[0]; a.y = ap[1];
      // B 4x16 f32: VGPR v holds rows k0+v (lanes 0-15) / k0+v+2 (lanes 16-31)
      float2 bw = W[2 * ks + half][nB];
      v2f b; b.x = bw.x; b.y = bw.y;
      if (ks & 1)
        acc1 = __builtin_amdgcn_wmma_f32_16x16x4_f32(false, a, false, b,
                                                     (short)0, acc1, false, false);
      else
        acc0 = __builtin_amdgcn_wmma_f32_16x16x4_f32(false, a, false, b,
                                                     (short)0, acc0, false, false);
    }
    v8f acc = acc0 + acc1;
#pragma unroll
    for (int v = 0; v < 8; ++v)          // C/D: VGPR v -> M = v + 8*half
      sG[v + half * 8][nB] = acc[v];
  };

  // elementwise LSTM cell: consumes sG, updates c and h (h written to one or
  // two destinations at column offset 64 / 0 respectively)
  auto cell_elem = [&](const float* bias, float (*C)[kD],
                       float (*self_in)[kKP], float (*next_in)[kKP]) {
    for (int e = tid; e < kMT * kHid; e += kThreads) {
      int r = e / kHid, c = e - r * kHid;
      float gi = sG[r][c]          + bias[c];
      float gf = sG[r][64 + c]     + bias[64 + c];
      float gg = sG[r][128 + c]    + bias[128 + c];
      float go = sG[r][192 + c]    + bias[192 + c];
      float i_ = sigf(gi), f_ = sigf(gf), o_ = sigf(go);
      float g_ = tanh_fast(gg);
      float cn = f_ * C[r][c] + i_ * g_;
      C[r][c] = cn;
      float h = o_ * tanh_fast(cn);
      self_in[r][64 + c] = h;            // recurrent input for this layer
      if (next_in) next_in[r][c] = h;    // feed next layer
    }
  };

  // proj[16 x 64] = h1[16 x ~52] @ Wout^T ; waves 0..3 (uniform branch -> EXEC all 1s)
  auto proj_wmma = [&]() {
    if (wv < 4) {
      v8f acc0 = {0.f, 0.f, 0.f, 0.f, 0.f, 0.f, 0.f, 0.f};
      v8f acc1 = {0.f, 0.f, 0.f, 0.f, 0.f, 0.f, 0.f, 0.f};
#pragma unroll
      for (int ks = 0; ks < kKSo; ++ks) {
        const float* ap = &sHH1[rA][64 + 4 * ks + 2 * half];
        v2f a; a.x = ap[0]; a.y = ap[1];
        float2 bw = sWo[2 * ks + half][nB];
        v2f b; b.x = bw.x; b.y = bw.y;
        if (ks & 1)
          acc1 = __builtin_amdgcn_wmma_f32_16x16x4_f32(false, a, false, b,
                                                       (short)0, acc1, false, false);
        else
          acc0 = __builtin_amdgcn_wmma_f32_16x16x4_f32(false, a, false, b,
                                                       (short)0, acc0, false, false);
      }
      v8f acc = acc0 + acc1;
#pragma unroll
      for (int v = 0; v < 8; ++v) sG[v + half * 8][nB] = acc[v];
    }
  };

  auto proj_update = [&](int tglob) {   // x <- x + proj + bout ; emit to out
    for (int i = tid; i < kMT * kD; i += kThreads) {
      int r = i >> 6, d = i & 63;
      float xn = sXH0[r][d] + sG[r][d] + sBo[d];
      sXH0[r][d] = xn;
      out[((long)(bm + r) * kTout + tglob) * kD + d] = xn;
    }
  };

  auto two_layer = [&]() {
    gates_wmma(sXH0, sW0); __syncthreads();
    cell_elem(sB0, sC0, sXH0, sHH1); __syncthreads();
    gates_wmma(sHH1, sW1); __syncthreads();
    cell_elem(sB1, sC1, sHH1, nullptr); __syncthreads();
  };

  // ---------------- warm-up over x_init (also copies x_init -> out) ----------------
  for (int t = 0; t < kTin; ++t) {
    const int buf = t & 1;
    asm volatile("s_wait_asynccnt 0x0" ::: "memory");  // staged x for step t landed
    __syncthreads();
    if (t + 1 < kTin) stage_issue(t + 1, buf ^ 1);     // prefetch next step
    {
      float2 v = *(const float2*)&sXst[buf][xr][xc];
      *(float2*)&sXH0[xr][xc] = v;
      *(float2*)&out[((long)(bm + xr) * kTout + t) * kD + xc] = v;
    }
    __syncthreads();
    two_layer();
  }

  // ---------------- autoregressive steps ----------------
  for (int s = 0; s < kTar; ++s) {
    if (s > 0) two_layer();          // cell on current x (written by prev proj_update)
    proj_wmma(); __syncthreads();
    proj_update(kTin + s); __syncthreads();
  }
}

extern "C" void kernel_launch(void* const* d_in, const int* in_sizes, int n_in,
                              void* d_out, int out_size, void* d_ws, size_t ws_size,
                              hipStream_t stream) {
  (void)in_sizes; (void)n_in; (void)out_size; (void)d_ws; (void)ws_size;
  const float* x    = (const float*)d_in[0];
  const float* Wih0 = (const float*)d_in[1];
  const float* Whh0 = (const float*)d_in[2];
  const float* bih0 = (const float*)d_in[3];
  const float* bhh0 = (const float*)d_in[4];
  const float* Wih1 = (const float*)d_in[5];
  const float* Whh1 = (const float*)d_in[6];
  const float* bih1 = (const float*)d_in[7];
  const float* bhh1 = (const float*)d_in[8];
  const float* Wout = (const float*)d_in[9];
  const float* bout = (const float*)d_in[10];
  float* out = (float*)d_out;

  lstm_persistent_kernel<<<dim3(kBS / kMT), dim3(kThreads), 0, stream>>>(
      x, Wih0, Whh0, bih0, bhh0, Wih1, Whh1, bih1, bhh1, Wout, bout, out);
}